// ConvSov_22814866276878
// MI455X (gfx1250) — compile-verified
//
#include <hip/hip_runtime.h>
#include <hip/hip_bf16.h>
#include <math.h>

typedef __attribute__((ext_vector_type(2))) float v2f;
typedef __attribute__((ext_vector_type(8))) float v8f;

#define Bsz 4
#define NIN 32
#define Hdim 14
#define Wdim 14
#define DIN 8
#define NOUT 32
#define DOUT 16
#define OH 12
#define OW 12
#define PROJ 512
#define NPIX (Bsz*NIN*Hdim*Wdim)   // 25088
#define NSITE (Bsz*OH*OW)          // 576

// ---------------------------------------------------------------------------
// fp32 WMMA GEMM, compile-time shapes: C[NPIX,N] = act(A[NPIX,K] @ W + bias)
// block = 256 thr = 8 waves. Block tile: 128 rows x 128 cols.
// Per wave: 16-wide N slice, 8 M-subtiles -> 8 WMMAs per B fragment.
// K processed in pairs via permuted K order: one ds_load_b128 feeds 2 WMMAs
// (logical blocks {k8,k8+1,k8+4,k8+5} and {k8+2,k8+3,k8+6,k8+7}).
// A tile staged in LDS, row stride 68 (16B-aligned float4, conflict-free).
// ---------------------------------------------------------------------------
template<int N, int K, int RELU>
__global__ __launch_bounds__(256) void gemm_f32_wmma(
    const float* __restrict__ A, const float* __restrict__ W,
    const float* __restrict__ bias, float* __restrict__ C)
{
    constexpr int KC = (K < 64) ? K : 64;        // k-chunk held in LDS
    __shared__ float As[128 * 68];               // 34.8 KB

    const int lane = threadIdx.x & 31;
    const int wave = threadIdx.x >> 5;
    const int m0 = blockIdx.x * 128;
    const int n0 = blockIdx.y * 128 + wave * 16;

    const int half = lane >> 4;                  // 0: lanes 0-15, 1: 16-31
    const int mr   = lane & 15;

    v8f acc[8];
    #pragma unroll
    for (int t = 0; t < 8; ++t) acc[t] = (v8f){0.f,0.f,0.f,0.f,0.f,0.f,0.f,0.f};

    #pragma unroll 1
    for (int kc = 0; kc < K; kc += KC) {
        __syncthreads();
        // stage A: 128 rows x KC cols, float4 chunks
        #pragma unroll
        for (int t = 0; t < (128 * KC / 4) / 256; ++t) {
            const int idx = threadIdx.x + t * 256;
            const int m = idx / (KC / 4);
            const int k = (idx - m * (KC / 4)) * 4;
            const float4 val =
                *(const float4*)(A + (size_t)(m0 + m) * K + kc + k);
            *(float4*)(&As[m * 68 + k]) = val;
        }
        // prefetch next A chunk while this one is consumed
        if (K > KC && kc + KC < K) {
            const char* nxt = (const char*)(A + (size_t)m0 * K + kc + KC)
                              + (size_t)threadIdx.x * 128;
            __builtin_prefetch(nxt, 0, 3);
        }
        __syncthreads();

        #pragma unroll
        for (int k8 = 0; k8 < KC; k8 += 8) {
            // B rows for the two permuted logical K-blocks
            const float* wp =
                W + (size_t)(kc + k8 + 4 * half) * N + n0 + mr;
            v2f b1, b2;
            b1.x = wp[0];            // rows k8   | k8+4
            b1.y = wp[N];            // rows k8+1 | k8+5
            b2.x = wp[2 * N];        // rows k8+2 | k8+6
            b2.y = wp[3 * N];        // rows k8+3 | k8+7
            #pragma unroll
            for (int t = 0; t < 8; ++t) {
                const float4 av = *(const float4*)(
                    &As[(t * 16 + mr) * 68 + k8 + 4 * half]);
                v2f a1; a1.x = av.x; a1.y = av.y;
                v2f a2; a2.x = av.z; a2.y = av.w;
                acc[t] = __builtin_amdgcn_wmma_f32_16x16x4_f32(
                    false, a1, false, b1, (short)0, acc[t], false, false);
                acc[t] = __builtin_amdgcn_wmma_f32_16x16x4_f32(
                    false, a2, false, b2, (short)0, acc[t], false, false);
            }
        }
    }

    // C/D layout: VGPR r -> M = r + 8*half, N = n0 + (lane&15)
    const int n = n0 + mr;
    const float bv = bias[n];
    #pragma unroll
    for (int t = 0; t < 8; ++t) {
        #pragma unroll
        for (int r = 0; r < 8; ++r) {
            float v = acc[t][r] + bv;
            if (RELU) v = fmaxf(v, 0.f);
            C[(size_t)(m0 + t * 16 + r + 8 * half) * N + n] = v;
        }
    }
}

// ---------------------------------------------------------------------------
// Dynamic routing: one block (8 waves) per (b, oh, ow) site.
// lane = output capsule j (32 = wave width); each wave owns 36 input capsules.
// ---------------------------------------------------------------------------
__device__ __forceinline__ float wave_max(float x) {
    #pragma unroll
    for (int off = 16; off > 0; off >>= 1) x = fmaxf(x, __shfl_xor(x, off, 32));
    return x;
}
__device__ __forceinline__ float wave_sum(float x) {
    #pragma unroll
    for (int off = 16; off > 0; off >>= 1) x += __shfl_xor(x, off, 32);
    return x;
}

__global__ __launch_bounds__(256) void routing_kernel(
    const float* __restrict__ U, float* __restrict__ outV,
    float* __restrict__ outEnt)
{
    __shared__ float bb[288 * 32];       // 36 KB: routing logits
    __shared__ float ps[8][32 * 16];     // 16 KB: per-wave partial s
    __shared__ float vsh[32 * 16];       //  2 KB: broadcast v

    const int site = blockIdx.x;
    const int b  = site / (OH * OW);
    const int rr = site - b * OH * OW;
    const int oh = rr / OW;
    const int ow = rr - oh * OW;
    const int j    = threadIdx.x & 31;   // output capsule
    const int wave = threadIdx.x >> 5;   // 0..7
    const int i0 = wave * 36;

    float v[16], s[16], ur[16];

    #pragma unroll 1
    for (int it = 0; it < 3; ++it) {
        #pragma unroll
        for (int d = 0; d < 16; ++d) s[d] = 0.f;

        #pragma unroll 1
        for (int i = i0; i < i0 + 36; ++i) {
            const int nin = i / 9, kk = i - nin * 9, k1 = kk / 3, k2 = kk - k1 * 3;
            const int pix = ((b * NIN + nin) * Hdim + oh + k1) * Wdim + ow + k2;
            const float4* up = (const float4*)(U + (size_t)pix * PROJ + j * 16);
            float4 a0 = up[0], a1 = up[1], a2 = up[2], a3 = up[3];
            ur[0]=a0.x; ur[1]=a0.y; ur[2]=a0.z; ur[3]=a0.w;
            ur[4]=a1.x; ur[5]=a1.y; ur[6]=a1.z; ur[7]=a1.w;
            ur[8]=a2.x; ur[9]=a2.y; ur[10]=a2.z; ur[11]=a2.w;
            ur[12]=a3.x; ur[13]=a3.y; ur[14]=a3.z; ur[15]=a3.w;

            float c;
            if (it == 0) {
                c = 1.f / 32.f;                   // softmax(0) is uniform
            } else {
                float dotv = 0.f;
                #pragma unroll
                for (int d = 0; d < 16; ++d) dotv += ur[d] * v[d];
                const float bi = (it == 1) ? dotv : (bb[i * 32 + j] + dotv);
                bb[i * 32 + j] = bi;
                const float mx = wave_max(bi);
                const float e  = __expf(bi - mx);
                const float sm = wave_sum(e);
                c = e / sm;
                if (it == 2) {
                    const float t = wave_sum(-c * __logf(c + 1e-16f));
                    if (j == 0)
                        outEnt[(size_t)b * 288 * (OH * OW)
                               + (size_t)i * (OH * OW) + oh * OW + ow] =
                            t / logf((float)NOUT);
                }
            }
            #pragma unroll
            for (int d = 0; d < 16; ++d) s[d] += c * ur[d];
        }

        // cross-wave reduction of s, then squash -> v
        #pragma unroll
        for (int d = 0; d < 16; ++d) ps[wave][j * 16 + d] = s[d];
        __syncthreads();
        if (wave == 0) {
            float tot[16];
            #pragma unroll
            for (int d = 0; d < 16; ++d) {
                float acc = 0.f;
                #pragma unroll
                for (int w = 0; w < 8; ++w) acc += ps[w][j * 16 + d];
                tot[d] = acc;
            }
            float ns = 0.f;
            #pragma unroll
            for (int d = 0; d < 16; ++d) ns += tot[d] * tot[d];
            const float f = ns / (1.f + ns) * rsqrtf(ns + 1e-16f);
            #pragma unroll
            for (int d = 0; d < 16; ++d) vsh[j * 16 + d] = f * tot[d];
        }
        __syncthreads();
        #pragma unroll
        for (int d = 0; d < 16; ++d) v[d] = vsh[j * 16 + d];
    }

    // write v_j: (B, NOUT, oH, oW, DOUT)
    if (wave == 0) {
        float* vo = outV
            + (((size_t)(b * NOUT + j) * (OH * OW)) + oh * OW + ow) * DOUT;
        #pragma unroll
        for (int d = 0; d < 16; ++d) vo[d] = v[d];
    }
}

extern "C" void kernel_launch(void* const* d_in, const int* in_sizes, int n_in,
                              void* d_out, int out_size, void* d_ws, size_t ws_size,
                              hipStream_t stream) {
    const float* x  = (const float*)d_in[0];
    const float* W1 = (const float*)d_in[1];
    const float* b1 = (const float*)d_in[2];
    const float* W2 = (const float*)d_in[3];
    const float* b2 = (const float*)d_in[4];
    const float* W3 = (const float*)d_in[5];
    const float* b3 = (const float*)d_in[6];
    float* out = (float*)d_out;

    float* H1 = (float*)d_ws;                         // 25088 x 512
    float* H2 = H1 + (size_t)NPIX * PROJ;             // 25088 x 128
    float* U  = H2 + (size_t)NPIX * 128;              // 25088 x 512

    // layer 1: x(25088x8) @ W1(8x512) + b1, relu
    gemm_f32_wmma<512, 8, 1><<<dim3(NPIX / 128, 4), 256, 0, stream>>>(x, W1, b1, H1);
    // layer 2: H1 @ W2(512x128) + b2, relu
    gemm_f32_wmma<128, 512, 1><<<dim3(NPIX / 128, 1), 256, 0, stream>>>(H1, W2, b2, H2);
    // layer 3: H2 @ W3(128x512) + b3
    gemm_f32_wmma<512, 128, 0><<<dim3(NPIX / 128, 4), 256, 0, stream>>>(H2, W3, b3, U);

    // routing: 8 waves per (b, oh, ow) site
    routing_kernel<<<NSITE, 256, 0, stream>>>(
        U, out, out + (size_t)Bsz * NOUT * OH * OW * DOUT);
}